// SSDTargetGenerator_36567351558161
// MI455X (gfx1250) — compile-verified
//
#include <hip/hip_runtime.h>
#include <stdint.h>

typedef __attribute__((ext_vector_type(2))) float v2f;
typedef __attribute__((ext_vector_type(8))) float v8f;

#define IOU_THRESH 0.5f
#define BIP_THRESH 1e-12f

// fmaxf(x,x) == canonicalize; its result is known-canonical to the backend,
// so downstream minnum/maxnum users skip the implicit re-canonicalization.
__device__ __forceinline__ float canon(float x) { return fmaxf(x, x); }

__device__ __forceinline__ unsigned long long pack_best(float v, int r) {
    // v >= 0 always (IoU), so raw float bits are monotonic.
    // Complement the row so ties prefer the SMALLEST row index under max().
    return (((unsigned long long)__float_as_uint(v)) << 32) |
           (unsigned long long)(0xFFFFFFFFu - (unsigned)r);
}

// ---------------------------------------------------------------------------
// Kernel 1: IoU matrix (COLUMN-major: iou[c*Nt + r]), one wave per 16-anchor
// tile. area_a[n]+area_b[m] computed on the matrix pipe via
// V_WMMA_F32_16X16X4_F32. GT boxes broadcast to LDS via gfx1250 async
// global->LDS copies (ASYNCcnt). LDS GT arrays padded to 128 rows (zero
// boxes) and the row dimension padded to Nt so the tile loop has ZERO
// divergent guards: per lane the 8 results are 8 consecutive rows of one
// column -> two unconditional global_store_b128.
// ---------------------------------------------------------------------------
__global__ __launch_bounds__(32) void iou_tile_kernel(
    const float* __restrict__ anchors,   // (N,4) center format
    const float* __restrict__ gt,        // (M,4) corner format
    float* __restrict__ iou,             // (Mp, Nt) column-major scratch
    int N, int M, int Nt) {
    __shared__ float4 gt_c[128];
    __shared__ float  gt_area[128];
    __shared__ float4 an_c[16];
    __shared__ float  an_area[16];

    const int lane = threadIdx.x;
    const int rowbase = blockIdx.x * 16;

    // --- async copy of GT boxes (16B rows) into LDS; zero-fill the pad ------
    for (int k = 0; k < 4; ++k) {
        int r = lane + 32 * k;
        if (r < M) {
            unsigned lds_off = (unsigned)(uintptr_t)(&gt_c[r]);
            unsigned voff    = (unsigned)(r * 16);
            asm volatile("global_load_async_to_lds_b128 %0, %1, %2"
                         :: "v"(lds_off), "v"(voff), "s"(gt)
                         : "memory");
        } else {
            gt_c[r]    = make_float4(0.0f, 0.0f, 0.0f, 0.0f);
            gt_area[r] = 0.0f;
        }
    }

    // --- anchor tile: center -> corner, area (row clamp makes pad rows ------
    //     replicate anchor N-1; those rows are never read downstream) --------
    if (lane < 16) {
        int r  = rowbase + lane;
        int rr = r < N ? r : N - 1;
        float4 a = ((const float4*)anchors)[rr];
        float hw = 0.5f * a.z, hh = 0.5f * a.w;
        an_c[lane]    = make_float4(a.x - hw, a.y - hh, a.x + hw, a.y + hh);
        an_area[lane] = a.z * a.w;
    }

    asm volatile("s_wait_asynccnt 0" ::: "memory");
    __syncthreads();

    for (int r = lane; r < M; r += 32)
        gt_area[r] = (gt_c[r].z - gt_c[r].x) * (gt_c[r].w - gt_c[r].y);
    __syncthreads();

    const int half = lane >> 4;   // 0: lanes 0-15, 1: lanes 16-31
    const int l16  = lane & 15;

    // A fragment (16x4 f32, M x K): lanes 0-15 hold K0,K1; lanes 16-31 K2,K3.
    // Column 0 = area_a, column 1 = ones, columns 2,3 = zero.
    v2f afrag;
    afrag[0] = half ? 0.0f : an_area[l16];
    afrag[1] = half ? 0.0f : 1.0f;

    // Hoist + pre-canonicalize the 8 anchor corner rows this lane needs
    // (tile-invariant -> canonicalize ONCE, not once per tile iteration).
    float4 ac[8];
#pragma unroll
    for (int v = 0; v < 8; ++v) {
        float4 t = an_c[v + half * 8];
        ac[v] = make_float4(canon(t.x), canon(t.y), canon(t.z), canon(t.w));
    }

    // This lane's 8 results are rows [rowbase + half*8, +8) of column `col`.
    float* outbase = iou + rowbase + half * 8;

    const int ntiles = (M + 15) / 16;   // cols padded in LDS up to 128
    for (int t = 0; t < ntiles; ++t) {
        int col  = t * 16 + l16;        // always < 128: unconditional LDS read
        float ab = gt_area[col];

        // B fragment (4x16 f32, K x N), mirrored layout of A:
        // row 0 = ones, row 1 = area_b, rows 2,3 = zero.
        v2f bfrag;
        bfrag[0] = half ? 0.0f : 1.0f;
        bfrag[1] = half ? 0.0f : ab;

        v8f acc = {};
        // D[n][m] = area_a[n] + area_b[m] on the matrix pipe
        acc = __builtin_amdgcn_wmma_f32_16x16x4_f32(
            /*neg_a=*/false, afrag, /*neg_b=*/false, bfrag,
            /*c_mod=*/(short)0, acc, /*reuse_a=*/false, /*reuse_b=*/false);

        float4 g = gt_c[col];           // single ds_load_b128
        // canonicalize GT components once per tile (not once per use)
        float gx1 = canon(g.x), gy1 = canon(g.y);
        float gx2 = canon(g.z), gy2 = canon(g.w);

        float val[8];
#pragma unroll
        for (int v = 0; v < 8; ++v) {
            float iw = fminf(ac[v].z, gx2) - fmaxf(ac[v].x, gx1);
            float ih = fminf(ac[v].w, gy2) - fmaxf(ac[v].y, gy1);
            iw = fmaxf(iw, 0.0f);
            ih = fmaxf(ih, 0.0f);
            float inter = iw * ih;
            float uni   = acc[v] - inter;   // area_a + area_b - inter
            val[v] = inter / uni;
        }
        // Two vectorized stores; pad columns (col >= M) land in the [M, Mp) column
        // pad region of the scratch matrix and are never read back.
        float4* dst = (float4*)(outbase + (size_t)col * Nt);
        dst[0] = make_float4(val[0], val[1], val[2], val[3]);
        dst[1] = make_float4(val[4], val[5], val[6], val[7]);
    }
}

// ---------------------------------------------------------------------------
// Kernel 1b: per-anchor max IoU + first-argmax (MaximumMatcher).
// Column-major: for fixed c, adjacent threads read adjacent addresses.
// ---------------------------------------------------------------------------
__global__ void rowmax_kernel(const float* __restrict__ iou,
                              float* __restrict__ rowmax,
                              int* __restrict__ rowarg,
                              int N, int M, int Nt) {
    int n = blockIdx.x * blockDim.x + threadIdx.x;
    if (n >= N) return;
    float best = -1.0f;
    int arg = 0;
    for (int c = 0; c < M; ++c) {
        float v = iou[(size_t)c * Nt + n];
        if (v > best) { best = v; arg = c; }
    }
    rowmax[n] = best;
    rowarg[n] = arg;
}

// ---------------------------------------------------------------------------
// Kernel 2a: per-column max over non-removed rows (L2-resident rescan,
// fully coalesced in column-major layout)
// ---------------------------------------------------------------------------
__global__ __launch_bounds__(256) void colmax_kernel(
    const float* __restrict__ iou,
    const unsigned* __restrict__ row_used,
    const unsigned* __restrict__ col_used,
    unsigned long long* __restrict__ colbest,
    int N, int M, int Nt) {
    int c = blockIdx.x;
    if (c >= M) return;
    __shared__ unsigned long long sred[256];
    unsigned long long best = 0ull;
    if (!col_used[c]) {
        const float* colp = iou + (size_t)c * Nt;
        for (int r = threadIdx.x; r < N; r += 256) {
            if (row_used[r]) continue;
            unsigned long long p = pack_best(colp[r], r);
            if (p > best) best = p;
        }
    }
    sred[threadIdx.x] = best;
    __syncthreads();
    for (int s = 128; s > 0; s >>= 1) {
        if (threadIdx.x < s && sred[threadIdx.x + s] > sred[threadIdx.x])
            sred[threadIdx.x] = sred[threadIdx.x + s];
        __syncthreads();
    }
    if (threadIdx.x == 0) colbest[c] = sred[0];
}

// ---------------------------------------------------------------------------
// Kernel 2b: single-block global pick (one greedy bipartite round)
// ---------------------------------------------------------------------------
__global__ __launch_bounds__(128) void pick_kernel(
    const unsigned long long* __restrict__ colbest,
    int* __restrict__ bip_match,
    unsigned* __restrict__ row_used,
    unsigned* __restrict__ col_used,
    unsigned* __restrict__ done, int M) {
    if (*done) return;
    __shared__ unsigned long long sv[128];
    __shared__ int sc[128];
    int t = threadIdx.x;
    unsigned long long p = 0ull;
    int c = -1;
    if (t < M && !col_used[t]) { p = colbest[t]; c = t; }
    sv[t] = p;
    sc[t] = c;
    __syncthreads();
    for (int s = 64; s > 0; s >>= 1) {
        if (t < s && sv[t + s] > sv[t]) { sv[t] = sv[t + s]; sc[t] = sc[t + s]; }
        __syncthreads();
    }
    if (t == 0) {
        unsigned long long bp = sv[0];
        float val = __uint_as_float((unsigned)(bp >> 32));
        if (val > BIP_THRESH) {
            int r  = (int)(0xFFFFFFFFu - (unsigned)(bp & 0xFFFFFFFFull));
            int c0 = sc[0];
            bip_match[r] = c0;
            row_used[r]  = 1u;
            col_used[c0] = 1u;
        } else {
            *done = 1u;
        }
    }
}

// ---------------------------------------------------------------------------
// Kernel 3: composite match + NaiveSampler + class/box encoding
// ---------------------------------------------------------------------------
__global__ void encode_kernel(const float* __restrict__ anchors,
                              const float* __restrict__ gt,
                              const int* __restrict__ gt_ids,
                              const int* __restrict__ bip_match,
                              const float* __restrict__ rowmax,
                              const int* __restrict__ rowarg,
                              float* __restrict__ out, int N) {
    int n = blockIdx.x * blockDim.x + threadIdx.x;
    if (n >= N) return;
    int m = bip_match[n];
    if (m < 0) m = (rowmax[n] >= IOU_THRESH) ? rowarg[n] : -1;
    bool pos = (m >= 0);
    int safe = pos ? m : 0;

    out[n] = pos ? ((float)gt_ids[safe] + 1.0f) : 0.0f;

    float4 a  = ((const float4*)anchors)[n];   // (cx, cy, w, h)
    float4 g4 = ((const float4*)gt)[safe];     // corners
    float gw  = g4.z - g4.x, gh = g4.w - g4.y;
    float gcx = g4.x + 0.5f * gw, gcy = g4.y + 0.5f * gh;
    float t0 = (gcx - a.x) / a.z * 10.0f;      // / 0.1
    float t1 = (gcy - a.y) / a.w * 10.0f;
    float t2 = __logf(gw / a.z) * 5.0f;        // / 0.2
    float t3 = __logf(gh / a.w) * 5.0f;

    float* bt = out + N;
    float* bm = out + N + 4 * (size_t)N;
    size_t o = (size_t)n * 4;
    bt[o + 0] = pos ? t0 : 0.0f;
    bt[o + 1] = pos ? t1 : 0.0f;
    bt[o + 2] = pos ? t2 : 0.0f;
    bt[o + 3] = pos ? t3 : 0.0f;
    float mk = pos ? 1.0f : 0.0f;
    bm[o + 0] = mk; bm[o + 1] = mk; bm[o + 2] = mk; bm[o + 3] = mk;
}

__global__ void init_kernel(int* bip_match, unsigned* row_used,
                            unsigned* col_used, unsigned* done,
                            unsigned long long* colbest, int N) {
    int n = blockIdx.x * blockDim.x + threadIdx.x;
    if (n < N) { bip_match[n] = -1; row_used[n] = 0u; }
    if (n < 128) { col_used[n] = 0u; colbest[n] = 0ull; }
    if (n == 0) *done = 0u;
}

extern "C" void kernel_launch(void* const* d_in, const int* in_sizes, int n_in,
                              void* d_out, int out_size, void* d_ws, size_t ws_size,
                              hipStream_t stream) {
    const float* anchors = (const float*)d_in[0];
    const float* gt      = (const float*)d_in[1];
    const int*   gt_ids  = (const int*)d_in[2];
    const int N  = in_sizes[0] / 4;
    const int M  = in_sizes[1] / 4;
    const int Mp = ((M + 15) / 16) * 16;   // padded column count
    const int Nt = ((N + 15) / 16) * 16;   // padded row count

    char* ws = (char*)d_ws;
    size_t off = 0;
    float* iou = (float*)(ws + off);            off += (size_t)Mp * Nt * sizeof(float);
    float* rowmax = (float*)(ws + off);         off += (size_t)N * sizeof(float);
    int* rowarg = (int*)(ws + off);             off += (size_t)N * sizeof(int);
    int* bip = (int*)(ws + off);                off += (size_t)N * sizeof(int);
    unsigned* row_used = (unsigned*)(ws + off); off += (size_t)N * sizeof(unsigned);
    unsigned* col_used = (unsigned*)(ws + off); off += 128 * sizeof(unsigned);
    off = (off + 7) & ~(size_t)7;
    unsigned long long* colbest = (unsigned long long*)(ws + off); off += 128 * 8;
    unsigned* done = (unsigned*)(ws + off);     off += sizeof(unsigned);

    init_kernel<<<(N + 255) / 256, 256, 0, stream>>>(bip, row_used, col_used,
                                                     done, colbest, N);
    iou_tile_kernel<<<Nt / 16, 32, 0, stream>>>(anchors, gt, iou, N, M, Nt);
    rowmax_kernel<<<(N + 255) / 256, 256, 0, stream>>>(iou, rowmax, rowarg, N, M, Nt);
    for (int i = 0; i < M; ++i) {
        colmax_kernel<<<M, 256, 0, stream>>>(iou, row_used, col_used, colbest, N, M, Nt);
        pick_kernel<<<1, 128, 0, stream>>>(colbest, bip, row_used, col_used, done, M);
    }
    encode_kernel<<<(N + 255) / 256, 256, 0, stream>>>(anchors, gt, gt_ids, bip,
                                                       rowmax, rowarg,
                                                       (float*)d_out, N);
}